// CvxpyProjectionLayer_73203422593450
// MI455X (gfx1250) — compile-verified
//
#include <hip/hip_runtime.h>

// Problem constants (match reference)
#define NACT   80
#define MCON   85
#define SOFTC  4
#define NHARD  81      // MCON - SOFTC
#define NZ     84      // NACT + SOFTC
#define MG     169     // MCON + SOFTC + NACT
#define ST     85      // LDS row stride (odd -> conflict-free strided access)
#define RHO_C  10.0f
#define SIGMA_C 1e-6f
#define PEN_C  1000.0f
#define ADMM_ITERS 200
#define BLK    256

typedef float v2f __attribute__((ext_vector_type(2)));
typedef float v8f __attribute__((ext_vector_type(8)));

typedef __attribute__((address_space(1))) int gint_t;
typedef __attribute__((address_space(3))) int lint_t;

// LDS layout (floats)
#define SA_OFF   0
#define SK_OFF   (SA_OFF + MCON*ST)     // normalized A: 85 x 80 (stride 85)
#define SKI_OFF  (SK_OFF + NZ*ST)       // K then L (Cholesky in place)
#define SH_OFF   (SKI_OFF + NZ*ST)      // K^{-1}
#define SQ_OFF   (SH_OFF + MG)
#define SZ_OFF   (SQ_OFF + NZ)
#define SW_OFF   (SZ_OFF + NZ)
#define SY_OFF   (SW_OFF + MG)
#define SU_OFF   (SY_OFF + MG)
#define SGZ_OFF  (SU_OFF + MG)
#define SRHS_OFF (SGZ_OFF + MG)
#define SMEM_FLOATS (SRHS_OFF + NZ)     // = 22602 floats = 90408 bytes

__global__ __launch_bounds__(BLK) void qp_admm_kernel(
    const float* __restrict__ x_raw, const float* __restrict__ Ag,
    const float* __restrict__ bg,    const float* __restrict__ lowg,
    float* __restrict__ out)
{
  extern __shared__ float smem[];
  float* sA   = smem + SA_OFF;
  float* sK   = smem + SK_OFF;
  float* sKI  = smem + SKI_OFF;
  float* sh   = smem + SH_OFF;
  float* sq   = smem + SQ_OFF;
  float* sz   = smem + SZ_OFF;
  float* sw   = smem + SW_OFF;
  float* sy   = smem + SY_OFF;
  float* su   = smem + SU_OFF;
  float* sgz  = smem + SGZ_OFF;
  float* srhs = smem + SRHS_OFF;

  const int tid  = threadIdx.x;
  const int samp = blockIdx.x;
  const float* Asamp = Ag + (size_t)samp * (MCON * NACT);

  // ---------------- Phase 0: stage A (async global->LDS), init state ------
#if __has_builtin(__builtin_amdgcn_global_load_async_to_lds_b32)
  for (int f = tid; f < MCON * NACT; f += BLK) {
    int r = f / NACT, c = f - r * NACT;
    __builtin_amdgcn_global_load_async_to_lds_b32(
        (gint_t*)(Asamp + f),
        (lint_t*)&sA[r * ST + c], 0, 0);
  }
#else
  for (int f = tid; f < MCON * NACT; f += BLK) {
    int r = f / NACT, c = f - r * NACT;
    sA[r * ST + c] = Asamp[f];
  }
#endif
  // overlap scalar state init with in-flight async copies
  for (int f = tid; f < MG; f += BLK) { sw[f] = 0.f; sy[f] = 0.f; }
  if (tid < NZ) sz[tid] = 0.f;
  if (tid < NACT) {
    sq[tid] = -x_raw[(size_t)samp * NACT + tid];
    sh[MCON + SOFTC + tid] = -lowg[(size_t)samp * NACT + tid];
  } else if (tid < NZ) {
    sq[tid] = 0.f;
  }
  if (tid >= MCON && tid < MCON + SOFTC) sh[tid] = 0.f;
#if __has_builtin(__builtin_amdgcn_global_load_async_to_lds_b32)
  #if __has_builtin(__builtin_amdgcn_s_wait_asynccnt)
  __builtin_amdgcn_s_wait_asynccnt(0);
  #else
  asm volatile("s_wait_asynccnt 0" ::: "memory");
  #endif
#endif
  __syncthreads();

  // ---------------- Phase 1: row-normalize A, build h -----------------------
  if (tid < MCON) {
    float ss = 0.f;
    for (int c = 0; c < NACT; ++c) { float v = sA[tid * ST + c]; ss += v * v; }
    float rn  = fmaxf(sqrtf(ss), 1e-12f);
    float inv = 1.0f / rn;
    for (int c = 0; c < NACT; ++c) sA[tid * ST + c] *= inv;
    sh[tid] = bg[(size_t)samp * MCON + tid] * inv;
  }
  __syncthreads();

  // ---------------- Phase 2: K_xx = rho*A^T A via V_WMMA_F32_16X16X4_F32 ----
  {
    const int wave = tid >> 5, lane = tid & 31;
    const int m = lane & 15, kb2 = (lane >> 4) << 1, mh = lane >> 4;
    for (int t = wave; t < 25; t += (BLK / 32)) {        // wave-uniform: EXEC all-1s
      const int ti = t / 5, tj = t - 5 * ti;
      v8f acc = {0.f, 0.f, 0.f, 0.f, 0.f, 0.f, 0.f, 0.f};
      for (int kb = 0; kb < 21; ++kb) {                  // k = 0..83
        int r0 = kb * 4 + kb2;
        v2f av, bv;
        av.x = sA[r0 * ST + ti * 16 + m];
        av.y = sA[(r0 + 1) * ST + ti * 16 + m];
        bv.x = sA[r0 * ST + tj * 16 + m];
        bv.y = sA[(r0 + 1) * ST + tj * 16 + m];
        acc = __builtin_amdgcn_wmma_f32_16x16x4_f32(false, av, false, bv,
                                                    (short)0, acc, false, false);
      }
      { // tail k-step: only k == 84 is real; rows 85..87 are zero
        v2f av = {0.f, 0.f}, bv = {0.f, 0.f};
        if (kb2 == 0) {
          av.x = sA[84 * ST + ti * 16 + m];
          bv.x = sA[84 * ST + tj * 16 + m];
        }
        acc = __builtin_amdgcn_wmma_f32_16x16x4_f32(false, av, false, bv,
                                                    (short)0, acc, false, false);
      }
#pragma unroll
      for (int v = 0; v < 8; ++v)
        sK[(ti * 16 + mh * 8 + v) * ST + tj * 16 + m] = RHO_C * acc[v];
    }
  }
  __syncthreads();

  // ---------------- Phase 3: K borders + diagonal ---------------------------
  for (int f = tid; f < SOFTC * NACT; f += BLK) {        // K_xs = -rho*A_soft^T
    int t = f / NACT, c = f - t * NACT;
    float val = -RHO_C * sA[(NHARD + t) * ST + c];
    sK[c * ST + (NACT + t)] = val;
    sK[(NACT + t) * ST + c] = val;
  }
  if (tid < 16) {                                        // K_ss
    int t = tid >> 2, t2 = tid & 3;
    sK[(NACT + t) * ST + NACT + t2] =
        (t == t2) ? (2.f * RHO_C + 2.f * PEN_C + SIGMA_C) : 0.f;
  }
  if (tid < NACT)                                        // + rho*I + (1+sigma)I
    sK[tid * ST + tid] += RHO_C + 1.f + SIGMA_C;
  __syncthreads();

  // ---------------- Phase 4: in-place Cholesky K = L L^T --------------------
  for (int k = 0; k < NZ; ++k) {
    if (tid == 0) sK[k * ST + k] = sqrtf(sK[k * ST + k]);
    __syncthreads();
    float dinv = 1.0f / sK[k * ST + k];
    for (int i = k + 1 + tid; i < NZ; i += BLK) sK[i * ST + k] *= dinv;
    __syncthreads();
    int rem = NZ - 1 - k;
    for (int f = tid; f < rem * rem; f += BLK) {
      int j = k + 1 + f / rem, i = k + 1 + f % rem;
      if (i >= j) sK[i * ST + j] -= sK[i * ST + k] * sK[j * ST + k];
    }
    __syncthreads();
  }

  // ---------------- Phase 5: K^{-1} = L^{-T} L^{-1} (one column per thread) -
  if (tid < NZ) {
    const int j = tid;
    for (int i = 0; i < NZ; ++i) {                       // forward: L^{-1} e_j
      float acc = (i == j) ? 1.f : 0.f;
      for (int k = j; k < i; ++k) acc -= sK[i * ST + k] * sKI[k * ST + j];
      sKI[i * ST + j] = (i < j) ? 0.f : acc / sK[i * ST + i];
    }
    for (int i = NZ - 1; i >= 0; --i) {                  // backward: L^{-T} col
      float acc = sKI[i * ST + j];
      for (int k = i + 1; k < NZ; ++k) acc -= sK[k * ST + i] * sKI[k * ST + j];
      sKI[i * ST + j] = acc / sK[i * ST + i];
    }
  }
  __syncthreads();

  // ---------------- Phase 6: 200 ADMM iterations ----------------------------
  for (int it = 0; it < ADMM_ITERS; ++it) {
    if (tid < MG) su[tid] = RHO_C * sw[tid] - sy[tid];   // u = rho*w - y
    __syncthreads();

    // rhs = sigma*z - q + G^T u   (2 threads per output, shfl pair-reduce)
    if (tid < 2 * NACT) {
      int c = tid >> 1, half = tid & 1;
      int r0 = half ? 43 : 0, r1 = half ? MCON : 43;
      float acc = 0.f;
      for (int r = r0; r < r1; ++r) acc += sA[r * ST + c] * su[r];
      acc += __shfl_xor(acc, 1, 32);
      if (!half) srhs[c] = SIGMA_C * sz[c] - sq[c] + acc - su[MCON + SOFTC + c];
    } else if (tid < 2 * NACT + SOFTC) {
      int t = tid - 2 * NACT;
      srhs[NACT + t] = SIGMA_C * sz[NACT + t] - su[NHARD + t] - su[MCON + t];
    }
    __syncthreads();

    // z = K^{-1} rhs
    if (tid < 2 * NZ) {
      int i = tid >> 1, half = tid & 1;
      int k0 = half * 42;
      float acc = 0.f;
      for (int k = k0; k < k0 + 42; ++k) acc += sKI[i * ST + k] * srhs[k];
      acc += __shfl_xor(acc, 1, 32);
      if (!half) sz[i] = acc;
    }
    __syncthreads();

    // Gz (structured)
    if (tid < 2 * MCON) {
      int r = tid >> 1, half = tid & 1;
      int c0 = half * 40;
      float acc = 0.f;
      for (int c = c0; c < c0 + 40; ++c) acc += sA[r * ST + c] * sz[c];
      acc += __shfl_xor(acc, 1, 32);
      if (!half) {
        float g = acc;
        if (r >= NHARD) g -= sz[NACT + (r - NHARD)];     // A_soft x - s
        sgz[r] = g;
      }
    } else if (tid < 2 * MCON + SOFTC) {
      int t = tid - 2 * MCON;
      sgz[MCON + t] = -sz[NACT + t];                     // -s
    } else if (tid < 2 * MCON + SOFTC + NACT) {
      int c = tid - (2 * MCON + SOFTC);
      sgz[MCON + SOFTC + c] = -sz[c];                    // -x
    }
    __syncthreads();

    // w = min(Gz + y/rho, h);  y += rho*(Gz - w)
    if (tid < MG) {
      float g  = sgz[tid];
      float wn = fminf(g + sy[tid] * (1.0f / RHO_C), sh[tid]);
      sw[tid] = wn;
      sy[tid] = sy[tid] + RHO_C * (g - wn);
    }
    __syncthreads();
  }

  if (tid < NACT) out[(size_t)samp * NACT + tid] = sz[tid];
}

extern "C" void kernel_launch(void* const* d_in, const int* in_sizes, int n_in,
                              void* d_out, int out_size, void* d_ws, size_t ws_size,
                              hipStream_t stream) {
  const float* x_raw = (const float*)d_in[0];   // [B, 80]
  const float* A     = (const float*)d_in[1];   // [B, 85, 80]
  const float* b     = (const float*)d_in[2];   // [B, 85]
  const float* lower = (const float*)d_in[3];   // [B, 80]
  float* out = (float*)d_out;                   // [B, 80]
  (void)d_ws; (void)ws_size; (void)n_in;

  const int B = in_sizes[0] / NACT;             // 1024
  const size_t shmem = (size_t)SMEM_FLOATS * sizeof(float);  // ~90 KB (< 320 KB/WGP)
  qp_admm_kernel<<<dim3(B), dim3(BLK), shmem, stream>>>(x_raw, A, b, lower, out);
}